// RankingLossFunc_72284299592092
// MI455X (gfx1250) — compile-verified
//
#include <hip/hip_runtime.h>
#include <hip/hip_bf16.h>

// Ranking-loss collapse for MI455X (gfx1250, wave32).
//
// loss = N(N-1)*DELTA - N*sum_i diag_i + u.v,
//   u = sum_i X_i/||X_i||, v = sum_j Y_j/||Y_j||, diag_i = (X_i.Y_i)/max(||X_i||*||Y_i||, EPS)
//
// O(N*D), bandwidth-bound: 50.3 MB of fp32 reads -> ~2.2us floor at 23.3 TB/s.
// Per-row reductions done on the matrix unit via V_WMMA_F32_16X16X4_F32 Gram diagonals.

#define NROWS 8192
#define DDIM  768
#define DELTA 0.2
#define EPSV  1e-8f

typedef float v2f __attribute__((ext_vector_type(2)));
typedef float v8f __attribute__((ext_vector_type(8)));

#define KC       64                    // K-chunk staged in LDS per wave
#define LSTRIDE  (KC + 4)              // 68 dwords; 68 % 64 == 4 -> WMMA feed hits all 64 banks
#define WAVES    4
#define RPW      16                    // rows per wave (WMMA M dim)
#define RPB      (WAVES * RPW)         // 64 rows per block

// ws layout (floats): [0,768)=u  [768,1536)=v  [1536]=diagSum
#define WS_FLOATS (2 * DDIM + 1)

__device__ __forceinline__ float pick8(v8f c, int idx) {
    // dynamic extract from the 8-VGPR accumulator via cndmask chain (no scratch)
    float r = c[0];
    r = (idx == 1) ? c[1] : r;
    r = (idx == 2) ? c[2] : r;
    r = (idx == 3) ? c[3] : r;
    r = (idx == 4) ? c[4] : r;
    r = (idx == 5) ? c[5] : r;
    r = (idx == 6) ? c[6] : r;
    r = (idx == 7) ? c[7] : r;
    return r;
}

__global__ void __launch_bounds__(128) rank_zero_ws(float* __restrict__ W) {
    int i = blockIdx.x * blockDim.x + threadIdx.x;
    if (i < WS_FLOATS) W[i] = 0.0f;
}

__global__ void __launch_bounds__(128) rank_stats_kernel(
    const float* __restrict__ X, const float* __restrict__ Y, float* __restrict__ W)
{
    __shared__ float lX[WAVES][RPW * LSTRIDE];   // 4*4352B
    __shared__ float lY[WAVES][RPW * LSTRIDE];   // 4*4352B
    __shared__ float stat[WAVES][3][RPW];        // sxx/syy/sxy then diag
    __shared__ float inva[WAVES][2][RPW];        // 1/||X_i||, 1/||Y_i||
    __shared__ float ublk[DDIM];
    __shared__ float vblk[DDIM];

    const int tid  = threadIdx.x;
    const int w    = tid >> 5;
    const int lane = tid & 31;
    const int rowbase = blockIdx.x * RPB + w * RPW;

    for (int d = tid; d < DDIM; d += blockDim.x) { ublk[d] = 0.0f; vblk[d] = 0.0f; }
    __syncthreads();

    // ---- pass 1: WMMA Gram diagonals over D ------------------------------
    v8f cxx = {}; v8f cyy = {}; v8f cxy = {};
    const int m     = lane & 15;             // matrix row within tile
    const int khalf = (lane >> 4) << 1;      // lanes 16..31 hold K+2,K+3

    for (int kc = 0; kc < DDIM; kc += KC) {
        // stage 16 rows x 64 cols of X and Y: one coalesced 256B b64 load per row
        #pragma unroll 4
        for (int r = 0; r < RPW; ++r) {
            long off = (long)(rowbase + r) * DDIM + kc + 2 * lane;
            *(v2f*)&lX[w][r * LSTRIDE + 2 * lane] = *(const v2f*)(X + off);
            *(v2f*)&lY[w][r * LSTRIDE + 2 * lane] = *(const v2f*)(Y + off);
        }
        __syncthreads();
        #pragma unroll
        for (int ks = 0; ks < KC / 4; ++ks) {
            int k = ks * 4 + khalf;
            // A(16x4) layout == B(4x16) layout for Gram products: same LDS read
            v2f ax = *(const v2f*)&lX[w][m * LSTRIDE + k];
            v2f ay = *(const v2f*)&lY[w][m * LSTRIDE + k];
            cxx = __builtin_amdgcn_wmma_f32_16x16x4_f32(false, ax, false, ax, (short)0, cxx, false, false);
            cyy = __builtin_amdgcn_wmma_f32_16x16x4_f32(false, ay, false, ay, (short)0, cyy, false, false);
            cxy = __builtin_amdgcn_wmma_f32_16x16x4_f32(false, ax, false, ay, (short)0, cxy, false, false);
        }
        __syncthreads();
    }

    // ---- extract Gram diagonals (C/D layout: i<8 -> {vgpr=i,lane=i};
    //      i>=8 -> {vgpr=i-8, lane=i+16}) -----------------------------------
    {
        bool lo = (lane < 8), hi = (lane >= 24);
        if (lo || hi) {
            int r   = lo ? lane : (lane - 16);   // 0..7 or 8..15
            int idx = lo ? lane : (lane - 24);   // accumulator VGPR index
            stat[w][0][r] = pick8(cxx, idx);
            stat[w][1][r] = pick8(cyy, idx);
            stat[w][2][r] = pick8(cxy, idx);
        }
    }
    __syncthreads();

    if (lane < RPW) {
        float sxx = stat[w][0][lane];
        float syy = stat[w][1][lane];
        float sxy = stat[w][2][lane];
        float xn = sqrtf(sxx), yn = sqrtf(syy);
        float diag = sxy / fmaxf(xn * yn, EPSV);
        inva[w][0][lane] = (xn > 0.0f) ? 1.0f / xn : 0.0f;
        inva[w][1][lane] = (yn > 0.0f) ? 1.0f / yn : 0.0f;
        stat[w][0][lane] = diag;
    }
    __syncthreads();

    if (tid == 0) {
        float s = 0.0f;
        for (int ww = 0; ww < WAVES; ++ww)
            for (int r = 0; r < RPW; ++r) s += stat[ww][0][r];
        atomicAdd(&W[2 * DDIM], s);              // diagSum
    }

    // ---- pass 2: u,v accumulation (re-reads are L2-resident: 50MB << 192MB)
    for (int c = 0; c < DDIM / 32; ++c) {
        int d = c * 32 + lane;
        float fu = 0.0f, fv = 0.0f;
        #pragma unroll 4
        for (int r = 0; r < RPW; ++r) {
            long off = (long)(rowbase + r) * DDIM + d;   // coalesced 128B per row
            fu = fmaf(X[off], inva[w][0][r], fu);        // inva: LDS broadcast read
            fv = fmaf(Y[off], inva[w][1][r], fv);
        }
        atomicAdd(&ublk[d], fu);                 // ds_add_f32
        atomicAdd(&vblk[d], fv);
    }
    __syncthreads();
    for (int d = tid; d < DDIM; d += blockDim.x) {
        atomicAdd(&W[d], ublk[d]);               // global_atomic_add_f32
        atomicAdd(&W[DDIM + d], vblk[d]);
    }
}

__global__ void __launch_bounds__(256) rank_finalize_kernel(
    const float* __restrict__ W, float* __restrict__ out)
{
    __shared__ float red[256];
    int t = threadIdx.x;
    float s = 0.0f;
    for (int d = t; d < DDIM; d += 256) s += W[d] * W[DDIM + d];
    red[t] = s;
    __syncthreads();
    for (int k = 128; k > 0; k >>= 1) {
        if (t < k) red[t] += red[t + k];
        __syncthreads();
    }
    if (t == 0) {
        double diagSum = (double)W[2 * DDIM];
        double loss = (double)DELTA * (double)NROWS * (double)(NROWS - 1)
                      - (double)NROWS * diagSum
                      + (double)red[0];
        out[0] = (float)loss;
    }
}

extern "C" void kernel_launch(void* const* d_in, const int* in_sizes, int n_in,
                              void* d_out, int out_size, void* d_ws, size_t ws_size,
                              hipStream_t stream) {
    const float* X = (const float*)d_in[0];
    const float* Y = (const float*)d_in[1];
    float* W   = (float*)d_ws;
    float* out = (float*)d_out;

    rank_zero_ws<<<(WS_FLOATS + 127) / 128, 128, 0, stream>>>(W);
    rank_stats_kernel<<<NROWS / RPB, WAVES * 32, 0, stream>>>(X, Y, W);
    rank_finalize_kernel<<<1, 256, 0, stream>>>(W, out);
}